// CustomConv1D_d_82549271429534
// MI455X (gfx1250) — compile-verified
//
#include <hip/hip_runtime.h>
#include <hip/hip_bf16.h>
#include <math.h>

// Problem constants (match reference setup_inputs exactly)
#define TAPS   21
#define CCH    237
#define BATCH  32
#define TSTEPS 512                  // L / TAPS = 10752 / 21
#define ROWLEN (TAPS * CCH)         // 4977 contiguous floats per (b,t) window
#define TILE_T 16                   // t-rows per workgroup (matches WMMA M=16)

typedef __attribute__((ext_vector_type(2))) float v2f;
typedef __attribute__((ext_vector_type(8))) float v8f;

__global__ __launch_bounds__(256)
void CustomConv1D_d_82549271429534_kernel(const float* __restrict__ x,
                                          const float* __restrict__ p3,
                                          const float* __restrict__ p4,
                                          float* __restrict__ out) {
    __shared__ float w_lds[TAPS];
    __shared__ float partials[16][33];   // [row][lane], padded: conflict-free column reads
    __shared__ float rowsum[16];

    const int tid  = threadIdx.x;
    const int lane = tid & 31;
    const int wv   = tid >> 5;

    // ---- per-block softmax weights (nealmon), stable, f32-exact expf ----
    if (tid == 0) {
        const float a = p3[0], b = p4[0];
        float logits[TAPS];
        float mx = -INFINITY;
        #pragma unroll
        for (int i = 0; i < TAPS; ++i) {
            float fi = (float)(i + 1);
            logits[i] = a * fi + b * fi * fi;
            mx = fmaxf(mx, logits[i]);
        }
        float s = 0.0f;
        #pragma unroll
        for (int i = 0; i < TAPS; ++i) { float e = expf(logits[i] - mx); w_lds[i] = e; s += e; }
        float inv = 1.0f / s;
        #pragma unroll
        for (int i = 0; i < TAPS; ++i) w_lds[i] *= inv;
    }
    __syncthreads();

    // ---- block geometry ----
    const int b    = blockIdx.x >> 5;         // 32 tiles per batch row
    const int t0   = (blockIdx.x & 31) * TILE_T;
    const float* src = x + ((size_t)b * (TSTEPS * TAPS) + (size_t)t0 * TAPS) * CCH;

    // ---- stage 1: streaming weighted partial sums (HBM-bound part) ----
    // Wave wv handles rows 2wv and 2wv+1; lane-consecutive coalesced loads.
    #pragma unroll
    for (int rr = 0; rr < 2; ++rr) {
        const int row = wv * 2 + rr;
        const float* p = src + (size_t)row * ROWLEN;
        float acc = 0.0f;
        for (int k = 0; k < TAPS; ++k) {
            const float wk = w_lds[k];
            const float* pk = p + k * CCH;
            for (int c = lane; c < CCH; c += 32)
                acc = fmaf(pk[c], wk, acc);
        }
        partials[row][lane] = acc;
    }
    __syncthreads();

    // ---- stage 2: 16x32 lane-partial matrix times ones-vector via
    //      8 chained V_WMMA_F32_16X16X4_F32 (wave 0, EXEC all-ones) ----
    if (tid < 32) {
        const int m   = lane & 15;
        const int khi = (lane >> 4) << 1;     // lanes 0-15 hold K=kb+0/1; 16-31 hold K=kb+2/3
        v8f c = {};
        const v2f ones = {1.0f, 1.0f};        // B = 4x16 all-ones (layout-invariant)
        #pragma unroll
        for (int kb = 0; kb < 32; kb += 4) {
            v2f a;
            a.x = partials[m][kb + khi];
            a.y = partials[m][kb + khi + 1];
            // D[m][n] += sum_k A[m][k] * 1  -> row sums replicated across n
            c = __builtin_amdgcn_wmma_f32_16x16x4_f32(
                    false, a, false, ones, (short)0, c, false, false);
        }
        // D layout: VGPR r holds M=r (lanes 0-15) and M=r+8 (lanes 16-31); all n equal.
        if ((lane & 15) == 0) {
            const int hi = (lane >> 4) * 8;   // 0 for lane 0, 8 for lane 16
            #pragma unroll
            for (int r = 0; r < 8; ++r) rowsum[r + hi] = c[r];
        }
    }
    __syncthreads();

    // ---- stage 3: broadcast 16 scalars across 237 channels, coalesced ----
    float* obase = out + ((size_t)b * TSTEPS + t0) * CCH;
    for (int i = tid; i < TILE_T * CCH; i += 256)
        obase[i] = rowsum[i / CCH];
}

extern "C" void kernel_launch(void* const* d_in, const int* in_sizes, int n_in,
                              void* d_out, int out_size, void* d_ws, size_t ws_size,
                              hipStream_t stream) {
    (void)in_sizes; (void)n_in; (void)d_ws; (void)ws_size; (void)out_size;
    const float* x  = (const float*)d_in[0];
    const float* p3 = (const float*)d_in[1];
    const float* p4 = (const float*)d_in[2];
    float* out      = (float*)d_out;

    const int blocks = BATCH * (TSTEPS / TILE_T);   // 32 * 32 = 1024
    CustomConv1D_d_82549271429534_kernel<<<blocks, 256, 0, stream>>>(x, p3, p4, out);
}